// GCLSDA_Encoder_4956392260349
// MI455X (gfx1250) — compile-verified
//
#include <hip/hip_runtime.h>
#include <hip/hip_bf16.h>
#include <stdint.h>

// ---------------------------------------------------------------------------
// GCLSDA / LightGCN encoder for MI455X (gfx1250, wave32).
//   1) Build CSR from COO on device (histogram -> 2-level scan -> fill).
//   2) 3x { CSR SpMM (wave-per-row, L2-resident gathers, async-to-LDS
//           pipelined), fused perturb + accumulate + cl/final writes }.
// No float atomics anywhere.  All launches on `stream`; d2d copies via
// hipMemcpyAsync only (graph-capture safe).
// k_spmm is defined FIRST so the disasm snippet shows the hot loop.
// ---------------------------------------------------------------------------

#define USER_NUM 100000
#define ITEM_NUM 100000
#define NNODES   200000
#define EMB      64
#define NEDGES   6400000
#define NLAYERS  3
#define EPSV     0.1f
#define NP1      (NNODES + 1)
#define CHUNK    1024
#define NB       ((NP1 + CHUNK - 1) / CHUNK)   // 196 blocks of 1024
static_assert(NB <= 256, "single-block scan of partials requires NB<=256");

// CDNA5 async global->LDS path (ASYNCcnt). Builtin signature (probe-derived):
//   void __builtin_amdgcn_global_load_async_to_lds_b64(v2i AS1*, v2i AS3*,
//                                                      imm int off, imm cpol)
// Guarded so the host pass and older toolchains fall back to sync gathers.
#if __has_builtin(__builtin_amdgcn_global_load_async_to_lds_b64) && \
    __has_builtin(__builtin_amdgcn_s_wait_asynccnt)
#define HAVE_ASYNC_LDS 1
typedef int v2i_t __attribute__((vector_size(8)));
typedef __attribute__((address_space(1))) v2i_t as1_v2i;
typedef __attribute__((address_space(3))) v2i_t as3_v2i;
#else
#define HAVE_ASYNC_LDS 0
#endif

// ------------------------------- CSR SpMM ----------------------------------
// One wave per row. Each lane owns float2 of the 64-wide row. Edge metadata
// loaded 32-wide and broadcast via wave32 shuffles. Row gathers are pipelined
// through LDS with CDNA5 async loads (depth-3, 4 slots/wave) when available.
__global__ __launch_bounds__(256) void k_spmm(const unsigned* __restrict__ row_ptr,
                                              const unsigned* __restrict__ ecol,
                                              const float* __restrict__ evalv,
                                              const float* __restrict__ cur,
                                              float* __restrict__ nxt) {
  const int lane = threadIdx.x & 31;
  const int row  = (blockIdx.x * blockDim.x + threadIdx.x) >> 5;
  if (row >= NNODES) return;
  const unsigned s = row_ptr[row];
  const unsigned e = row_ptr[row + 1];
  float2 acc = make_float2(0.f, 0.f);

#if HAVE_ASYNC_LDS
  __shared__ __align__(16) char abuf[8][4][256];   // 8 waves x 4 slots x 256B
  char (*slot)[256] = abuf[threadIdx.x >> 5];
#endif

  for (unsigned b = s; b < e; b += 32) {
    const unsigned idx = b + (unsigned)lane;
    unsigned c = 0;
    float    v = 0.f;
    if (idx < e) {
      if (idx + 32 < e) __builtin_prefetch(ecol + idx + 32, 0, 0);
      c = ecol[idx];
      v = evalv[idx];
    }
    const int cnt = (int)min(32u, e - b);

#if HAVE_ASYNC_LDS
    // prologue: loads for edges 0..2 of this batch
#pragma unroll
    for (int d = 0; d < 3; ++d) {
      int jj = (d < cnt) ? d : (cnt - 1);
      unsigned cj = __shfl(c, jj, 32);
      const float* src = cur + (size_t)cj * EMB + lane * 2;
      __builtin_amdgcn_global_load_async_to_lds_b64(
          (as1_v2i*)(float*)src, (as3_v2i*)&slot[d][lane * 8], 0, 0);
    }
    for (int j = 0; j < cnt; ++j) {
      int jn = j + 3;
      if (jn >= cnt) jn = cnt - 1;          // pad to keep outstanding constant
      unsigned cj = __shfl(c, jn, 32);
      const float* src = cur + (size_t)cj * EMB + lane * 2;
      __builtin_amdgcn_global_load_async_to_lds_b64(
          (as1_v2i*)(float*)src, (as3_v2i*)&slot[(j + 3) & 3][lane * 8], 0, 0);
      __builtin_amdgcn_s_wait_asynccnt(3);  // oldest (edge j) is complete
      float vj = __shfl(v, j, 32);
      float2 x = *(const float2*)&slot[j & 3][lane * 8];
      acc.x = fmaf(vj, x.x, acc.x);
      acc.y = fmaf(vj, x.y, acc.y);
    }
    __builtin_amdgcn_s_wait_asynccnt(0);    // drain before slot reuse next batch
#else
    for (int j = 0; j < cnt; ++j) {
      unsigned cj = __shfl(c, j, 32);
      float    vj = __shfl(v, j, 32);
      const float2 x = *(const float2*)(cur + (size_t)cj * EMB + lane * 2);
      acc.x = fmaf(vj, x.x, acc.x);
      acc.y = fmaf(vj, x.y, acc.y);
    }
#endif
  }
  *(float2*)(nxt + (size_t)row * EMB + lane * 2) = acc;
}

// -------------------------------- utilities --------------------------------

__global__ __launch_bounds__(256) void k_zero_u32(unsigned* __restrict__ p, int n) {
  int i = blockIdx.x * blockDim.x + threadIdx.x;
  if (i < n) p[i] = 0u;
}

// counts[row]++ for every edge (u32 atomics, L2-resident counters)
__global__ __launch_bounds__(256) void k_hist(const long long* __restrict__ rows,
                                              unsigned* __restrict__ counts) {
  int i = blockIdx.x * blockDim.x + threadIdx.x;
  if (i < NEDGES) {
    if (i + 16384 < NEDGES) __builtin_prefetch(rows + i + 16384, 0, 0);
    atomicAdd(&counts[(int)rows[i]], 1u);
  }
}

// per-1024-chunk sums of counts -> partial[NB]
__global__ __launch_bounds__(256) void k_chunk_sums(const unsigned* __restrict__ counts,
                                                    unsigned* __restrict__ partial) {
  __shared__ unsigned sh[256];
  const int tid = threadIdx.x;
  const int base = blockIdx.x * CHUNK + tid * 4;
  unsigned t = 0;
#pragma unroll
  for (int k = 0; k < 4; ++k) {
    int i = base + k;
    t += (i < NP1) ? counts[i] : 0u;
  }
  sh[tid] = t;
  __syncthreads();
  for (int off = 128; off > 0; off >>= 1) {
    if (tid < off) sh[tid] += sh[tid + off];
    __syncthreads();
  }
  if (tid == 0) partial[blockIdx.x] = sh[0];
}

// single-block exclusive scan of partial[NB] (NB<=256), in place
__global__ __launch_bounds__(256) void k_scan_partial(unsigned* __restrict__ partial) {
  __shared__ unsigned sh[256];
  const int tid = threadIdx.x;
  unsigned v = (tid < NB) ? partial[tid] : 0u;
  sh[tid] = v;
  __syncthreads();
  for (int off = 1; off < 256; off <<= 1) {
    unsigned t = (tid >= off) ? sh[tid - off] : 0u;
    __syncthreads();
    sh[tid] += t;
    __syncthreads();
  }
  unsigned excl = (tid == 0) ? 0u : sh[tid - 1];
  if (tid < NB) partial[tid] = excl;
}

// exclusive scan of counts -> row_ptr (and cursor copy for the fill pass)
__global__ __launch_bounds__(256) void k_scan_final(const unsigned* __restrict__ counts,
                                                    const unsigned* __restrict__ partial,
                                                    unsigned* __restrict__ row_ptr,
                                                    unsigned* __restrict__ cursor) {
  __shared__ unsigned sh[256];
  const int tid = threadIdx.x;
  const int base = blockIdx.x * CHUNK + tid * 4;
  unsigned v[4];
  unsigned tsum = 0;
#pragma unroll
  for (int k = 0; k < 4; ++k) {
    int i = base + k;
    v[k] = (i < NP1) ? counts[i] : 0u;
    tsum += v[k];
  }
  sh[tid] = tsum;
  __syncthreads();
  for (int off = 1; off < 256; off <<= 1) {
    unsigned t = (tid >= off) ? sh[tid - off] : 0u;
    __syncthreads();
    sh[tid] += t;
    __syncthreads();
  }
  unsigned run = partial[blockIdx.x] + (sh[tid] - tsum);
#pragma unroll
  for (int k = 0; k < 4; ++k) {
    int i = base + k;
    if (i < NP1) {
      row_ptr[i] = run;
      if (i < NNODES) cursor[i] = run;
    }
    run += v[k];
  }
}

// scatter edges into CSR slots (u32 atomics on cursors)
__global__ __launch_bounds__(256) void k_fill(const long long* __restrict__ rows,
                                              const long long* __restrict__ cols,
                                              const float* __restrict__ vals,
                                              unsigned* __restrict__ cursor,
                                              unsigned* __restrict__ ecol,
                                              float* __restrict__ evalv) {
  int i = blockIdx.x * blockDim.x + threadIdx.x;
  if (i < NEDGES) {
    if (i + 16384 < NEDGES) {
      __builtin_prefetch(rows + i + 16384, 0, 0);
      __builtin_prefetch(cols + i + 16384, 0, 0);
      __builtin_prefetch(vals + i + 16384, 0, 0);
    }
    int r = (int)rows[i];
    unsigned pos = atomicAdd(&cursor[r], 1u);
    ecol[pos]  = (unsigned)cols[i];
    evalv[pos] = vals[i];
  }
}

// ---------------- fused perturb + accumulate + outputs ---------------------
// ego += sign(ego) * (noise_k / max(||noise_k||,1e-12)) * EPS    (in place)
// mode 0: acc  = ego; cl = ego
// mode 1: acc += ego
// mode 2: acc  = (acc + ego) / 3
__global__ __launch_bounds__(256) void k_perturb(float* __restrict__ ego,
                                                 const float* __restrict__ nz,
                                                 float* __restrict__ acc,
                                                 float* __restrict__ cl,
                                                 int mode) {
  const int lane = threadIdx.x & 31;
  const int node = (blockIdx.x * blockDim.x + threadIdx.x) >> 5;
  if (node >= NNODES) return;
  const size_t off = (size_t)node * EMB + lane * 2;

  float2 ev = *(const float2*)(ego + off);
  float2 nv = *(const float2*)(nz + off);

  float ss = nv.x * nv.x + nv.y * nv.y;
#pragma unroll
  for (int d = 16; d > 0; d >>= 1) ss += __shfl_xor(ss, d, 32);
  float nrm = fmaxf(sqrtf(ss), 1e-12f);
  float sc  = EPSV / nrm;

  float sgx = (float)((ev.x > 0.f) - (ev.x < 0.f));
  float sgy = (float)((ev.y > 0.f) - (ev.y < 0.f));
  float2 p;
  p.x = ev.x + sgx * nv.x * sc;
  p.y = ev.y + sgy * nv.y * sc;

  *(float2*)(ego + off) = p;   // becomes `cur` for the next layer

  if (mode == 0) {
    *(float2*)(acc + off) = p;
    *(float2*)(cl + off)  = p;
  } else if (mode == 1) {
    float2 a = *(const float2*)(acc + off);
    a.x += p.x; a.y += p.y;
    *(float2*)(acc + off) = a;
  } else {
    float2 a = *(const float2*)(acc + off);
    a.x = (a.x + p.x) / 3.0f;
    a.y = (a.y + p.y) / 3.0f;
    *(float2*)(acc + off) = a;
  }
}

// --------------------------------- launch ----------------------------------

extern "C" void kernel_launch(void* const* d_in, const int* in_sizes, int n_in,
                              void* d_out, int out_size, void* d_ws, size_t ws_size,
                              hipStream_t stream) {
  const float*      user_emb = (const float*)d_in[0];
  const float*      item_emb = (const float*)d_in[1];
  const long long*  rows     = (const long long*)d_in[2];
  const long long*  cols     = (const long long*)d_in[3];
  const float*      vals     = (const float*)d_in[4];
  const float*      noise    = (const float*)d_in[5];

  float* out_final = (float*)d_out;                       // [NNODES, 64]
  float* out_cl    = out_final + (size_t)NNODES * EMB;    // [NNODES, 64]

  // workspace carve (256B aligned slices); total ~156 MB
  char* w = (char*)d_ws;
  auto carve = [&](size_t bytes) -> char* {
    char* p = w;
    w += (bytes + 255) & ~(size_t)255;
    return p;
  };
  float*    curA    = (float*)carve((size_t)NNODES * EMB * sizeof(float));
  float*    curB    = (float*)carve((size_t)NNODES * EMB * sizeof(float));
  unsigned* ecol    = (unsigned*)carve((size_t)NEDGES * sizeof(unsigned));
  float*    evalv   = (float*)carve((size_t)NEDGES * sizeof(float));
  unsigned* counts  = (unsigned*)carve((size_t)NP1 * sizeof(unsigned));
  unsigned* row_ptr = (unsigned*)carve((size_t)NP1 * sizeof(unsigned));
  unsigned* cursor  = (unsigned*)carve((size_t)NNODES * sizeof(unsigned));
  unsigned* partial = (unsigned*)carve((size_t)NB * sizeof(unsigned));
  (void)ws_size; (void)in_sizes; (void)n_in; (void)out_size;

  const int edge_blocks = (NEDGES + 255) / 256;          // 25000
  const int node_blocks = (NNODES * 32 + 255) / 256;     // 25000 (wave per node)

  // ego_0 = concat(user_emb, item_emb)
  (void)hipMemcpyAsync(curA, user_emb, (size_t)USER_NUM * EMB * sizeof(float),
                       hipMemcpyDeviceToDevice, stream);
  (void)hipMemcpyAsync(curA + (size_t)USER_NUM * EMB, item_emb,
                       (size_t)ITEM_NUM * EMB * sizeof(float),
                       hipMemcpyDeviceToDevice, stream);

  // ---- build CSR (rebuilt every call: deterministic work, no caching) ----
  k_zero_u32<<<(NP1 + 255) / 256, 256, 0, stream>>>(counts, NP1);
  k_hist<<<edge_blocks, 256, 0, stream>>>(rows, counts);
  k_chunk_sums<<<NB, 256, 0, stream>>>(counts, partial);
  k_scan_partial<<<1, 256, 0, stream>>>(partial);
  k_scan_final<<<NB, 256, 0, stream>>>(counts, partial, row_ptr, cursor);
  k_fill<<<edge_blocks, 256, 0, stream>>>(rows, cols, vals, cursor, ecol, evalv);

  // ---- 3 propagation layers ----
  float* cur = curA;
  float* nxt = curB;
  for (int k = 0; k < NLAYERS; ++k) {
    k_spmm<<<node_blocks, 256, 0, stream>>>(row_ptr, ecol, evalv, cur, nxt);
    k_perturb<<<node_blocks, 256, 0, stream>>>(nxt, noise + (size_t)k * NNODES * EMB,
                                               out_final, out_cl, k);
    float* t = cur; cur = nxt; nxt = t;
  }
}